// Renderer_66228395704421
// MI455X (gfx1250) — compile-verified
//
#include <hip/hip_runtime.h>
#include <hip/hip_bf16.h>
#include <math.h>

// UNISURF-style ray marcher on gfx1250.
// Layer-1 of the 3->32->1 occupancy MLP runs on V_WMMA_F32_16X16X4_F32:
//   A (16x4) = W1^T chunk (constant), B (4x16) = 16 point coords, D = hidden acts.
// 16 rays per wave32; lanes n and n+16 co-own ray n (matches A/B/D half-lane layouts).

typedef float v2f __attribute__((ext_vector_type(2)));
typedef float v8f __attribute__((ext_vector_type(8)));

#define N_STEPS  32
#define N_SECANT 8

__global__ __launch_bounds__(256) void unisurf_wmma_kernel(
    const float* __restrict__ cam,   // [3]
    const float* __restrict__ rd,    // [P,3]
    const float* __restrict__ W1,    // [3,32] row-major
    const float* __restrict__ b1,    // [32]
    const float* __restrict__ W2,    // [32,1]
    const float* __restrict__ b2,    // [1]
    float* __restrict__ out,         // [2*P]  (d_pred | occ_surf)
    int P)
{
    const int  lane = threadIdx.x & 31;
    const int  wid  = threadIdx.x >> 5;
    const bool hi   = lane >= 16;
    const int  m    = lane & 15;

    int ray = blockIdx.x * 128 + wid * 16 + m;
    const bool valid = (ray < P);
    if (!valid) ray = P - 1;          // clamp: keeps EXEC full for WMMA

    // ---- constants ----
    const float cx = cam[0], cy = cam[1], cz = cam[2];
    const float b2v = b2[0];

    // A matrices (W1^T, K = {x,y,z,pad}); ISA A-layout:
    // lanes 0-15: VGPR0=K0, VGPR1=K1 ; lanes 16-31: VGPR0=K2, VGPR1=K3(=0)
    v2f a1, a2;
    if (!hi) {
        a1.x = W1[m];        a1.y = W1[32 + m];        // hidden m:  w_x, w_y
        a2.x = W1[16 + m];   a2.y = W1[48 + m];        // hidden m+16
    } else {
        a1.x = W1[64 + m];   a1.y = 0.0f;              // hidden m:  w_z, pad
        a2.x = W1[80 + m];   a2.y = 0.0f;              // hidden m+16
    }

    // Layer-2 constants arranged per D-matrix lane layout:
    // lane n (lo) sees hiddens {v, 16+v}; lane n+16 (hi) sees {8+v, 24+v}
    const int base1 = hi ? 8 : 0;
    const int base2 = hi ? 24 : 16;
    float b1A[8], w2A[8], b1B[8], w2B[8];
    #pragma unroll
    for (int v = 0; v < 8; ++v) {
        b1A[v] = b1[base1 + v];
        w2A[v] = W2[base1 + v];
        b1B[v] = b1[base2 + v];
        w2B[v] = W2[base2 + v];
    }

    // ---- ray setup ----
    float dx = rd[3 * ray + 0], dy = rd[3 * ray + 1], dz = rd[3 * ray + 2];
    float inv = rsqrtf(dx * dx + dy * dy + dz * dz);
    float rx = dx * inv, ry = dy * inv, rz = dz * inv;

    float rcd  = rx * cx + ry * cy + rz * cz;
    float c2   = cx * cx + cy * cy + cz * cz;
    float us   = rcd * rcd - (c2 - 1.0f);           // RADIUS = 1
    bool  mask = us > 0.0f;
    float sq   = sqrtf(mask ? us : 0.0f);
    float i1   = mask ? fmaxf(sq - rcd, 0.0f) : 0.0f;
    float farv = fmaxf(i1, 0.5f + 1e-6f);           // NEAR + EPS

    const v8f czero = {0.f, 0.f, 0.f, 0.f, 0.f, 0.f, 0.f, 0.f};

    // Full-wave MLP evaluation at depth dmv; returns sigmoid occupancy.
    auto evalOcc = [&](float dmv) -> float {
        float px = cx + rx * dmv;
        float py = cy + ry * dmv;
        float pz = cz + rz * dmv;
        v2f bm;                                   // B layout: lo lanes K={x,y}, hi lanes K={z,0}
        bm.x = hi ? pz : px;
        bm.y = hi ? 0.0f : py;
        v8f d1 = __builtin_amdgcn_wmma_f32_16x16x4_f32(
            false, a1, false, bm, (short)0, czero, false, false);
        v8f d2 = __builtin_amdgcn_wmma_f32_16x16x4_f32(
            false, a2, false, bm, (short)0, czero, false, false);
        float acc = 0.0f;
        #pragma unroll
        for (int v = 0; v < 8; ++v) {
            float h0 = d1[v] + b1A[v]; h0 = fmaxf(h0, 0.0f); acc = fmaf(h0, w2A[v], acc);
            float h1 = d2[v] + b1B[v]; h1 = fmaxf(h1, 0.0f); acc = fmaf(h1, w2B[v], acc);
        }
        acc += __shfl_xor(acc, 16, 32);           // combine co-lane half sums
        float pre = acc + b2v;
        return 1.0f / (1.0f + __expf(-pre));
    };

    // ---- coarse march: first free->occupied crossing ----
    float dl = 0.f, fl = 0.f, dh = 0.f, fh = 0.f;
    float dprev = 0.f, fprev = 0.f;
    bool foundc = false;
    for (int s = 0; s < N_STEPS; ++s) {
        float t = (float)s * (1.0f / (float)(N_STEPS - 1));
        float d = 0.5f * (1.0f - t) + farv * t;
        float f = evalOcc(d) - 0.5f;
        if (s >= 1) {
            bool cross = (fprev < 0.0f) && (f >= 0.0f);
            bool take  = (s == 1) || (cross && !foundc);   // s==1 seeds idx=0 defaults
            dl = take ? dprev : dl;
            fl = take ? fprev : fl;
            dh = take ? d     : dh;
            fh = take ? f     : fh;
            foundc = foundc || cross;
        }
        dprev = d; fprev = f;
    }
    const bool found = foundc && mask;

    // ---- secant refinement ----
    #pragma unroll 1
    for (int it = 0; it < N_SECANT; ++it) {
        float den = fh - fl;
        den = (fabsf(den) < 1e-6f) ? 1e-6f : den;
        float dm = dl - fl * (dh - dl) / den;
        float fm = evalOcc(dm) - 0.5f;
        bool neg = fm < 0.0f;
        dl = neg ? dm : dl;
        fl = neg ? fm : fl;
        dh = neg ? dh : dm;
        fh = neg ? fh : fm;
    }
    float den  = fh - fl;
    den        = (fabsf(den) < 1e-6f) ? 1e-6f : den;
    float dmid = dl - fl * (dh - dl) / den;
    float dpred = found ? dmid : 0.0f;

    // ---- surface occupancy ----
    float occs = evalOcc(dpred);
    occs = found ? occs : 0.0f;

    if (valid && !hi) {
        out[ray]     = dpred;
        out[P + ray] = occs;
    }
}

extern "C" void kernel_launch(void* const* d_in, const int* in_sizes, int n_in,
                              void* d_out, int out_size, void* d_ws, size_t ws_size,
                              hipStream_t stream) {
    const float* cam = (const float*)d_in[0];
    const float* rd  = (const float*)d_in[1];
    const float* W1  = (const float*)d_in[2];
    const float* b1  = (const float*)d_in[3];
    const float* W2  = (const float*)d_in[4];
    const float* b2  = (const float*)d_in[5];
    float* out = (float*)d_out;

    int P = in_sizes[1] / 3;                 // ray_directions is [B=1, P, 3]
    int raysPerBlock = 128;                  // 8 waves * 16 rays
    int grid = (P + raysPerBlock - 1) / raysPerBlock;
    unisurf_wmma_kernel<<<grid, 256, 0, stream>>>(cam, rd, W1, b1, W2, b2, out, P);
}